// MoERouter_41772851921369
// MI455X (gfx1250) — compile-verified
//
#include <hip/hip_runtime.h>

// MoE router, fused: logits GEMM (fp32 WMMA 16x16x4) + masked softmax + top-2.
// B=8, T=4096, D=1024, E=64, K=2.  32768 tokens total.
//
// CDNA5 paths used:
//   - v_wmma_f32_16x16x4_f32 (fp32 matrix core, preserves reference numerics)
//   - global_load_async_to_lds_b128 + s_wait_asynccnt (double-buffered staging)
//
// Outputs concatenated flat in d_out (floats):
//   [0,      65536)    expert_weights (tok,2)
//   [65536, 131072)    expert_indices (tok,2)  -- written as float values
//   [131072, 2228224)  router_logits  (tok,64)
//   [2228224, 4325376) router_probs   (tok,64)

typedef __attribute__((ext_vector_type(2))) float v2f;
typedef __attribute__((ext_vector_type(8))) float v8f;
typedef __attribute__((ext_vector_type(4))) float f4;
typedef __attribute__((ext_vector_type(4))) int   v4i;

constexpr int Dm      = 1024;        // d_model
constexpr int Em      = 64;          // experts
constexpr int TOKENS  = 8 * 4096;
constexpr int BLK_TOK = 128;         // tokens per block (8 waves x 16)
constexpr int KC      = 32;          // K chunk staged in LDS (double buffered)
constexpr int PITCH   = 36;          // staging pitch (dwords): conflict-free b64 frags
constexpr int LPITCH  = 68;          // logits pitch (dwords): conflict-free scatter

constexpr int XSZ = BLK_TOK * PITCH;     // 4608 floats per X buffer
constexpr int BUF = XSZ + Em * PITCH;    // 6912 floats per (X+W) buffer

#ifndef __has_builtin
#define __has_builtin(x) 0
#endif

#if __has_builtin(__builtin_amdgcn_global_load_async_to_lds_b128) && \
    __has_builtin(__builtin_amdgcn_s_wait_asynccnt)
// Signature (from hipcc diagnostic): param0 = v4i AS1* (global src),
// param1 = v4i AS3* (LDS dst), then imm offset, imm cpol.
#define ASYNC_CP(l, g)                                                        \
    __builtin_amdgcn_global_load_async_to_lds_b128(                           \
        (__attribute__((address_space(1))) v4i*)(size_t)(g),                  \
        (__attribute__((address_space(3))) v4i*)(l), 0, 0)
#define ASYNC_WAIT() __builtin_amdgcn_s_wait_asynccnt(0)
#else
// Fallback: synchronous float4 copy (global -> VGPR -> LDS); barriers still correct.
#define ASYNC_CP(l, g) (*(f4*)(l) = *(const f4*)(g))
#define ASYNC_WAIT() ((void)0)
#endif

__global__ __launch_bounds__(256) void moe_router_kernel(
    const float* __restrict__ x,      // (32768, 1024)
    const int*   __restrict__ x_mask, // (32768)
    const float* __restrict__ Wt,     // (64, 1024)  row-major [expert][d]
    float* __restrict__ w_out,        // (32768, 2)
    float* __restrict__ idx_out,      // (32768, 2) as float
    float* __restrict__ logit_out,    // (32768, 64)
    float* __restrict__ prob_out)     // (32768, 64)
{
    __shared__ float smem[2 * BUF];   // 55296 B

    const int tid   = threadIdx.x;
    const int lane  = tid & 31;
    const int wave  = tid >> 5;
    const int tok0  = blockIdx.x * BLK_TOK;

    const int lrow  = lane & 15;            // M (A) / N (B,C) index within tile
    const int khalf = (lane >> 4) * 2;      // K sub-offset: 0 (lanes 0-15), 2 (16-31)

    v8f c0 = {}, c1 = {}, c2 = {}, c3 = {}; // 16 tokens x experts {0-15,16-31,32-47,48-63}

    // ---- async stage of one (X 128xKC, W 64xKC) chunk; 16B per lane per op ----
    auto stage = [&](int kc, float* buf) {
        float* sx = buf;          // X: 128 x PITCH
        float* sw = buf + XSZ;    // W:  64 x PITCH
        const int j  = (tid & 7) * 4;        // col (float4)
        const int r0 = tid >> 3;             // row base, 0..31
        #pragma unroll
        for (int rr = 0; rr < 4; ++rr) {
            const int row = r0 + rr * 32;
            ASYNC_CP(sx + row * PITCH + j,
                     x + (size_t)(tok0 + row) * Dm + kc + j);
        }
        #pragma unroll
        for (int rr = 0; rr < 2; ++rr) {
            const int row = r0 + rr * 32;
            ASYNC_CP(sw + row * PITCH + j,
                     Wt + (size_t)row * Dm + kc + j);
        }
    };

    // prologue: fill buffer 0
    stage(0, smem);
    ASYNC_WAIT();
    __syncthreads();

    for (int kc = 0; kc < Dm; kc += KC) {
        const int cur = (kc / KC) & 1;
        if (kc + KC < Dm)                        // prefetch next chunk into other buffer
            stage(kc + KC, smem + (cur ^ 1) * BUF);

        // ---- WMMA compute on current buffer: wave = 16 tokens x 64 experts ----
        const float* xb = smem + cur * BUF;
        const float* wb = xb + XSZ;
        const float* aB = xb + (wave * 16 + lrow) * PITCH + khalf;
        const float* b0 = wb + ( 0 + lrow) * PITCH + khalf;
        const float* b1 = wb + (16 + lrow) * PITCH + khalf;
        const float* b2 = wb + (32 + lrow) * PITCH + khalf;
        const float* b3 = wb + (48 + lrow) * PITCH + khalf;
        #pragma unroll
        for (int kk = 0; kk < KC; kk += 4) {
            v2f a = *(const v2f*)(aB + kk);
            c0 = __builtin_amdgcn_wmma_f32_16x16x4_f32(false, a, false, *(const v2f*)(b0 + kk), (short)0, c0, false, false);
            c1 = __builtin_amdgcn_wmma_f32_16x16x4_f32(false, a, false, *(const v2f*)(b1 + kk), (short)0, c1, false, false);
            c2 = __builtin_amdgcn_wmma_f32_16x16x4_f32(false, a, false, *(const v2f*)(b2 + kk), (short)0, c2, false, false);
            c3 = __builtin_amdgcn_wmma_f32_16x16x4_f32(false, a, false, *(const v2f*)(b3 + kk), (short)0, c3, false, false);
        }

        ASYNC_WAIT();      // next chunk's LDS writes landed (overlapped with WMMAs)
        __syncthreads();   // + every wave done reading current buffer
    }

    // LDS reuse after GEMM: logits scratch + per-row stats alias the buffers
    float* const smL   = smem;                       // logits 128 x LPITCH
    float* const stMax = smem + BLK_TOK * LPITCH;
    float* const stInv = stMax + BLK_TOK;
    float* const stM   = stInv + BLK_TOK;

    // ---- scatter C tiles (logits) into LDS: row = token, col = expert ----
    {
        const int rowBase = wave * 16 + (lane >> 4) * 8;   // M = r (+8 for hi half)
        #pragma unroll
        for (int r = 0; r < 8; ++r) {
            float* dst = smL + (rowBase + r) * LPITCH + lrow;
            dst[ 0] = c0[r];
            dst[16] = c1[r];
            dst[32] = c2[r];
            dst[48] = c3[r];
        }
    }
    __syncthreads();

    // ---- per-token masked softmax stats + top-2 (one thread per token) ----
    if (tid < BLK_TOK) {
        const int token = tok0 + tid;
        const float m = (float)x_mask[token];
        const float* row = smL + tid * LPITCH;
        float mx = -3.402823466e38f;
        #pragma unroll 8
        for (int e = 0; e < Em; ++e) mx = fmaxf(mx, m * row[e]);
        float sum = 0.f;
        float m1 = -3.402823466e38f, m2 = -3.402823466e38f;
        int   i1 = -1, i2 = -1;
        #pragma unroll 8
        for (int e = 0; e < Em; ++e) {
            const float v = m * row[e];
            sum += __expf(v - mx);
            if (v > m1)      { m2 = m1; i2 = i1; m1 = v; i1 = e; }
            else if (v > m2) { m2 = v;  i2 = e; }
        }
        stMax[tid] = mx;
        stInv[tid] = 1.f / sum;
        stM[tid]   = m;
        const float e1 = __expf(m1 - mx), e2 = __expf(m2 - mx);
        const float inv2 = 1.f / (e1 + e2);
        float w1 = e1 * inv2, w2 = e2 * inv2;
        if (m == 0.f) { w1 = 0.f; w2 = 0.f; i1 = -1; i2 = -1; }
        w_out[(size_t)token * 2 + 0]   = w1;
        w_out[(size_t)token * 2 + 1]   = w2;
        idx_out[(size_t)token * 2 + 0] = (float)i1;
        idx_out[(size_t)token * 2 + 1] = (float)i2;
    }
    __syncthreads();

    // ---- cooperative coalesced float4 stores of logits & probs ----
    #pragma unroll
    for (int it = 0; it < 8; ++it) {
        const int i = tid + it * 256;          // 2048 float4s = 128 rows x 16
        const int r = i >> 4;
        const int j = (i & 15) * 4;
        const float m = stM[r], mx = stMax[r], inv = stInv[r];
        const float* src = smL + r * LPITCH + j;
        f4 lg, pb;
        lg.x = m * src[0]; lg.y = m * src[1]; lg.z = m * src[2]; lg.w = m * src[3];
        pb.x = m * __expf(lg.x - mx) * inv;
        pb.y = m * __expf(lg.y - mx) * inv;
        pb.z = m * __expf(lg.z - mx) * inv;
        pb.w = m * __expf(lg.w - mx) * inv;
        const size_t base = (size_t)(tok0 + r) * Em + j;
        *(f4*)(logit_out + base) = lg;
        *(f4*)(prob_out  + base) = pb;
    }
}

extern "C" void kernel_launch(void* const* d_in, const int* in_sizes, int n_in,
                              void* d_out, int out_size, void* d_ws, size_t ws_size,
                              hipStream_t stream) {
    const float* x      = (const float*)d_in[0];
    const int*   x_mask = (const int*)  d_in[1];
    const float* Wt     = (const float*)d_in[2];

    float* out    = (float*)d_out;
    float* w_out  = out;
    float* i_out  = out + (size_t)TOKENS * 2;
    float* l_out  = out + (size_t)TOKENS * 4;
    float* p_out  = l_out + (size_t)TOKENS * Em;

    dim3 grid(TOKENS / BLK_TOK);   // 256 blocks
    dim3 block(256);               // 8 wave32
    moe_router_kernel<<<grid, block, 0, stream>>>(x, x_mask, Wt,
                                                  w_out, i_out, l_out, p_out);
}